// ISI_loss_function_59193239274224
// MI455X (gfx1250) — compile-verified
//
#include <hip/hip_runtime.h>
#include <hip/hip_bf16.h>
#include <stdint.h>

#define B_TOTAL   65536
#define P_TOTAL   2048
#define NT        292          // number of t values (2*T, T=146)
#define TBASE     878          // mid - T
#define ROWS      8            // rows per TDM chunk (one per wave)
#define CHUNKS    8            // chunks per block -> 64 rows/block
#define NBLOCKS   1024         // 1024 * 64 = 65536 rows
#define INV_SQRT2 0.70710678118654752f

typedef unsigned int v4u __attribute__((ext_vector_type(4)));
typedef int          v8i __attribute__((ext_vector_type(8)));
typedef int          v4i __attribute__((ext_vector_type(4)));

// Issue a TDM load of a [rows x 2048] f32 tile (row-major, stride 2048) from
// global memory into LDS at byte offset lds_off. D# layout per CDNA5 ISA ch.8.
static __device__ __forceinline__ void tdm_load_tile(const float* gptr,
                                                     unsigned lds_off,
                                                     unsigned rows) {
  unsigned long long ga = (unsigned long long)(uintptr_t)gptr;
  v4u g0;
  g0.x = 1u;                                             // count=1, user mode
  g0.y = lds_off;                                        // lds_addr
  g0.z = (unsigned)(ga & 0xFFFFFFFFu);                   // global_addr[31:0]
  g0.w = (unsigned)((ga >> 32) & 0x1FFFFFFu) | (2u << 30); // addr[56:32], type=2
  v8i g1;
  g1[0] = (int)(2u << 16);                 // workgroup_mask=0, data_size=2 (4B)
  g1[1] = (int)((P_TOTAL & 0xFFFFu) << 16);          // tensor_dim0 bits[15:0]
  g1[2] = (int)(((P_TOTAL >> 16) & 0xFFFFu) | (rows << 16)); // dim0 hi | dim1 lo
  g1[3] = (int)((rows >> 16) | ((unsigned)P_TOTAL << 16));   // dim1 hi | tile_dim0
  g1[4] = (int)(rows & 0xFFFFu);           // tile_dim1 = rows, tile_dim2 = 0
  g1[5] = (int)P_TOTAL;                    // tensor_dim0_stride[31:0] = 2048
  g1[6] = 0;                               // stride hi bits, dim1_stride lo
  g1[7] = 0;
  v4i gz4 = (v4i)(0);
  v8i gz8 = (v8i)(0);
  __builtin_amdgcn_tensor_load_to_lds(g0, g1, gz4, gz4, gz8, 0);
}

static __device__ __forceinline__ unsigned lds_offset_of(const void* p) {
  // Flat shared-aperture address: low 32 bits are the workgroup-relative LDS
  // byte offset (CDNA5 ISA 10.2 aperture mapping).
  return (unsigned)(uintptr_t)p;
}

// ---------------- Kernel 1: ISI moments ----------------
__global__ __launch_bounds__(256) void isi_moments_kernel(
    const float* __restrict__ isi_ch,   // (B,6)
    const float* __restrict__ isi_sr,   // (B,6)
    const float* __restrict__ isi_si,   // (B,6)
    float* __restrict__ mom)            // out: m1r,m2r,m1i,m2i
{
  __shared__ float4 red[256];
  const int tid = threadIdx.x;
  float s1r = 0.f, s2r = 0.f, s1i = 0.f, s2i = 0.f;
  for (int i = tid; i < B_TOTAL; i += 256) {
    float c  = isi_ch[i * 6];
    float cr = c * isi_sr[i * 6];
    float ci = c * isi_si[i * 6];
    s1r += cr; s2r += cr * cr;
    s1i += ci; s2i += ci * ci;
  }
  red[tid] = make_float4(s1r, s2r, s1i, s2i);
  __syncthreads();
  for (int s = 128; s > 0; s >>= 1) {
    if (tid < s) {
      float4 a = red[tid], b = red[tid + s];
      red[tid] = make_float4(a.x + b.x, a.y + b.y, a.z + b.z, a.w + b.w);
    }
    __syncthreads();
  }
  if (tid == 0) {
    const float inv = 1.0f / (float)B_TOTAL;
    mom[0] = red[0].x * inv;  // m1r
    mom[1] = red[0].y * inv;  // m2r
    mom[2] = red[0].z * inv;  // m1i
    mom[3] = red[0].w * inv;  // m2i
  }
}

// ---------------- Kernel 2: main streaming pass (TDM double-buffered) ---------
__global__ __launch_bounds__(256) void isi_main_kernel(
    const float* __restrict__ dnn,      // (B, 2048)
    const float* __restrict__ bb,       // (B,)
    const float* __restrict__ hh,       // (B,)
    const float* __restrict__ xr,       // (B,1)
    const float* __restrict__ xi,       // (B,1)
    const float* __restrict__ noise,    // (B,1)
    const float* __restrict__ prob,     // (1, 2048)
    const float* __restrict__ mom,      // m1r,m2r,m1i,m2i
    float* __restrict__ partials)       // (NBLOCKS,)
{
  __shared__ float buf[2][ROWS * P_TOTAL];   // 2 x 64KB double buffer
  __shared__ float p_lds[NT];
  __shared__ float red[256];

  const int tid  = threadIdx.x;
  const int wave = tid >> 5;
  const int lane = tid & 31;
  const int row0 = blockIdx.x * (ROWS * CHUNKS);

  for (int t = tid; t < NT; t += 256) p_lds[t] = prob[TBASE + t];

  const float m1r = mom[0], m2r = mom[1], m1i = mom[2], m2i = mom[3];
  const float M2  = m2r + m2i;

  // Prologue: start chunk 0 into buffer 0.
  if (wave == 0) {
    tdm_load_tile(dnn + (size_t)row0 * P_TOTAL, lds_offset_of(&buf[0][0]), ROWS);
  }

  float acc = 0.f;
  for (int c = 0; c < CHUNKS; ++c) {
    const int cur = c & 1;
    if (wave == 0) {
      if (c + 1 < CHUNKS) {
        // Previous barrier guarantees everyone finished reading buf[1-cur].
        tdm_load_tile(dnn + (size_t)(row0 + (c + 1) * ROWS) * P_TOTAL,
                      lds_offset_of(&buf[1 - cur][0]), ROWS);
        __builtin_amdgcn_s_wait_tensorcnt(1);  // chunk c is complete
      } else {
        __builtin_amdgcn_s_wait_tensorcnt(0);  // last chunk complete
      }
    }
    __syncthreads();  // chunk c visible to the whole workgroup

    const int    row = row0 + c * ROWS + wave;
    const float* rp  = &buf[cur][wave * P_TOTAL];

    const float gb = bb[row], gh = hh[row];
    const float vxr = xr[row], vxi = xi[row];
    const float nz = INV_SQRT2 * noise[row];
    const float al = gb * gh * vxr, be = nz - vxr;   // A_r = al*d0 + be
    const float ga = gb * gh * vxi, de = nz - vxi;   // A_i = ga*d0 + de

    for (int t = lane; t < NT; t += 32) {
      const float* q = rp + TBASE + t;
      float d0 = q[0];
      float S  = q[-876] + q[-584] + q[-292] + q[292] + q[584] + q[876];
      float Ar = fmaf(al, d0, be);
      float Ai = fmaf(ga, d0, de);
      float bs = gb * S;
      float v  = Ar * Ar + Ai * Ai
               + 2.0f * bs * fmaf(Ar, m1r, Ai * m1i)
               + bs * bs * M2;
      acc = fmaf(p_lds[t], v, acc);
    }
    __syncthreads();  // all waves done with buf[cur] before it is refilled
  }

  // Deterministic block tree-reduction.
  red[tid] = acc;
  __syncthreads();
  for (int s = 128; s > 0; s >>= 1) {
    if (tid < s) red[tid] += red[tid + s];
    __syncthreads();
  }
  if (tid == 0) partials[blockIdx.x] = red[0];
}

// ---------------- Kernel 3: finalize ----------------
__global__ __launch_bounds__(256) void isi_final_kernel(
    const float* __restrict__ partials, float* __restrict__ out)
{
  __shared__ float red[256];
  const int tid = threadIdx.x;
  float s = 0.f;
  for (int i = tid; i < NBLOCKS; i += 256) s += partials[i];
  red[tid] = s;
  __syncthreads();
  for (int k = 128; k > 0; k >>= 1) {
    if (tid < k) red[tid] += red[tid + k];
    __syncthreads();
  }
  if (tid == 0) out[0] = red[0] * (1.0f / (float)B_TOTAL);
}

extern "C" void kernel_launch(void* const* d_in, const int* in_sizes, int n_in,
                              void* d_out, int out_size, void* d_ws, size_t ws_size,
                              hipStream_t stream) {
  (void)in_sizes; (void)n_in; (void)out_size; (void)ws_size;
  const float* dnn   = (const float*)d_in[0];
  const float* b     = (const float*)d_in[1];
  const float* h     = (const float*)d_in[2];
  const float* xr    = (const float*)d_in[3];
  const float* xi    = (const float*)d_in[4];
  const float* isr   = (const float*)d_in[5];
  const float* isi   = (const float*)d_in[6];
  const float* ich   = (const float*)d_in[7];
  const float* noise = (const float*)d_in[8];
  const float* prob  = (const float*)d_in[9];

  float* ws       = (float*)d_ws;
  float* mom      = ws;        // 4 floats (+4 pad)
  float* partials = ws + 8;    // NBLOCKS floats

  isi_moments_kernel<<<1, 256, 0, stream>>>(ich, isr, isi, mom);
  isi_main_kernel<<<NBLOCKS, 256, 0, stream>>>(dnn, b, h, xr, xi, noise, prob,
                                               mom, partials);
  isi_final_kernel<<<1, 256, 0, stream>>>(partials, (float*)d_out);
}